// GATNet_25821343384096
// MI455X (gfx1250) — compile-verified
//
#include <hip/hip_runtime.h>
#include <hip/hip_bf16.h>

// ---------------------------------------------------------------------------
// 2-layer GAT on MI455X (gfx1250).
// GEMMs via V_WMMA_F32_16X16X4_F32 (fp32-exact matrix pipe), register-blocked
// 16x64 per wave; compile-time K/N strides so the hot loop is pure
// load-clause + WMMA with constant offsets. Edge segment-softmax via
// ordered-uint atomicMax + float atomicAdd (L2-resident on MI455X's 192MB L2).
// ---------------------------------------------------------------------------

typedef __attribute__((ext_vector_type(2))) float v2f;
typedef __attribute__((ext_vector_type(8))) float v8f;

#define EMB    128
#define HID    256
#define REPR   64
#define HEADS  8
#define OUT1   32
#define NEG_SLOPE 0.2f

// ---- order-preserving float <-> uint mapping (for atomicMax on floats) ----
__device__ __forceinline__ unsigned flt2ord(float x) {
    unsigned u = __float_as_uint(x);
    return (u & 0x80000000u) ? ~u : (u | 0x80000000u);
}
__device__ __forceinline__ float ord2flt(unsigned v) {
    unsigned u = (v & 0x80000000u) ? (v ^ 0x80000000u) : ~v;
    return __uint_as_float(u);
}
#define ORD_NEG_INF 0x007FFFFFu   // flt2ord(-inf)

// ---------------------------------------------------------------------------
// Workspace init (graph-safe; runs every call).
// ---------------------------------------------------------------------------
__global__ void init_ws(unsigned* m1, float* s1, float* agg1,
                        unsigned* m2, float* s2, float* agg2, int Nn) {
    long i = (long)blockIdx.x * blockDim.x + threadIdx.x;
    if (i < (long)Nn * HEADS) { m1[i] = ORD_NEG_INF; s1[i] = 0.f; }
    if (i < Nn)               { m2[i] = ORD_NEG_INF; s2[i] = 0.f; }
    if (i < (long)Nn * HID)   agg1[i] = 0.f;
    if (i < (long)Nn * REPR)  agg2[i] = 0.f;
}

// ---------------------------------------------------------------------------
// fp32 GEMM via V_WMMA_F32_16X16X4_F32. One wave computes a 16 x (16*NT)
// strip of C. Compile-time K/NCOLS: all loop addresses are constant offsets
// from pointers bumped by constants; B fragments for all NT tiles are loaded
// as one clause before the WMMA group so waits stagger instead of serialize.
// A operand (16x4): lanes 0-15 -> M=lane, VGPR0/1 = K0/K1;
//                   lanes 16-31 -> M=lane-16, VGPR0/1 = K2/K3.
// B operand (4x16): lanes 0-15 hold K=0..1, lanes 16-31 hold K=2..3, N=lane%16.
// C/D: VGPR r -> M = r (+8 for upper lanes), N = lane%16.
// ---------------------------------------------------------------------------
template <int NT, int K, int NCOLS>
__global__ __launch_bounds__(128) void gemm_wmma_f32(
    const float* __restrict__ A, const float* __restrict__ B,
    float* __restrict__ C, int M) {
    constexpr int GROUPS = NCOLS / (16 * NT);
    int wave = (blockIdx.x * blockDim.x + threadIdx.x) >> 5;
    int lane = threadIdx.x & 31;
    int tm = wave / GROUPS;
    int tq = wave - tm * GROUPS;
    if (tm * 16 >= M) return;                    // wave-uniform

    int half = lane >> 4;                        // 0 or 1
    int l16  = lane & 15;
    int arow = tm * 16 + l16;
    if (arow >= M) arow = M - 1;                 // clamp keeps EXEC full
    // Per-lane base pointers; inner loop adds only compile-time constants.
    const float* Ap = A + (long)arow * K + half * 2;
    const float* Bp = B + tq * (16 * NT) + l16 + (long)(half * 2) * NCOLS;

    v8f acc[NT];
#pragma unroll
    for (int j = 0; j < NT; ++j) acc[j] = (v8f){};

#pragma unroll 2
    for (int k0 = 0; k0 < K; k0 += 4) {
        v2f a; a.x = Ap[0]; a.y = Ap[1];
        v2f b[NT];
#pragma unroll
        for (int j = 0; j < NT; ++j) {           // one load clause, NT pairs
            b[j].x = Bp[j * 16];
            b[j].y = Bp[j * 16 + NCOLS];
        }
#pragma unroll
        for (int j = 0; j < NT; ++j) {           // drain with staggered waits
            acc[j] = __builtin_amdgcn_wmma_f32_16x16x4_f32(
                /*neg_a=*/false, a, /*neg_b=*/false, b[j],
                /*c_mod=*/(short)0, acc[j], /*reuse_a=*/false, /*reuse_b=*/false);
        }
        Ap += 4;
        Bp += 4 * NCOLS;
    }
#pragma unroll
    for (int r = 0; r < 8; ++r) {
        int row = tm * 16 + r + half * 8;
        if (row < M) {
            float* Cp = C + (long)row * NCOLS + tq * (16 * NT) + l16;
#pragma unroll
            for (int j = 0; j < NT; ++j) Cp[j * 16] = acc[j][r];
        }
    }
}

// ---------------------------------------------------------------------------
// Per-node attention coefficients: a_src[n,h] = <h[n,h,:], att_src[h,:]>
// ---------------------------------------------------------------------------
template <int H, int D>
__global__ void att_coef(const float* __restrict__ h,
                         const float* __restrict__ att_s,
                         const float* __restrict__ att_d,
                         float* __restrict__ asrc, float* __restrict__ adst,
                         int Nn) {
    int i = blockIdx.x * blockDim.x + threadIdx.x;
    if (i >= Nn * H) return;
    int n = i / H, hh = i - n * H;
    const float* hp = h + (long)n * (H * D) + hh * D;
    float ss = 0.f, sd = 0.f;
#pragma unroll
    for (int k = 0; k < D; ++k) {
        float x = hp[k];
        ss = fmaf(x, att_s[hh * D + k], ss);
        sd = fmaf(x, att_d[hh * D + k], sd);
    }
    asrc[i] = ss;
    adst[i] = sd;
}

// ---------------------------------------------------------------------------
// Edge pass 1: segment max of leaky_relu(a_src[s]+a_dst[d]) over dst.
// ---------------------------------------------------------------------------
template <int H>
__global__ void edge_max(const int* __restrict__ src, const int* __restrict__ dst,
                         int E, int Nn,
                         const float* __restrict__ asrc, const float* __restrict__ adst,
                         unsigned* __restrict__ mord) {
    int i = blockIdx.x * blockDim.x + threadIdx.x;
    if (i >= (E + Nn) * H) return;
    int e = i / H, h = i - e * H;
    int s = (e < E) ? src[e] : (e - E);
    int d = (e < E) ? dst[e] : (e - E);
    float v = asrc[s * H + h] + adst[d * H + h];
    v = v > 0.f ? v : NEG_SLOPE * v;
    atomicMax(&mord[d * H + h], flt2ord(v));
}

// ---------------------------------------------------------------------------
// Edge pass 2: segment sum of exp(e - max).
// ---------------------------------------------------------------------------
template <int H>
__global__ void edge_sum(const int* __restrict__ src, const int* __restrict__ dst,
                         int E, int Nn,
                         const float* __restrict__ asrc, const float* __restrict__ adst,
                         const unsigned* __restrict__ mord, float* __restrict__ ssum) {
    int i = blockIdx.x * blockDim.x + threadIdx.x;
    if (i >= (E + Nn) * H) return;
    int e = i / H, h = i - e * H;
    int s = (e < E) ? src[e] : (e - E);
    int d = (e < E) ? dst[e] : (e - E);
    float v = asrc[s * H + h] + adst[d * H + h];
    v = v > 0.f ? v : NEG_SLOPE * v;
    float p = __expf(v - ord2flt(mord[d * H + h]));
    atomicAdd(&ssum[d * H + h], p);
}

// ---------------------------------------------------------------------------
// Edge pass 3: scatter h[src] * alpha into agg[dst]. One block per edge;
// alpha[H] computed once into LDS, H*D lanes stream the message.
// ---------------------------------------------------------------------------
template <int H, int D>
__global__ __launch_bounds__(H * D) void edge_msg(
    const int* __restrict__ src, const int* __restrict__ dst, int E, int Nn,
    const float* __restrict__ asrc, const float* __restrict__ adst,
    const unsigned* __restrict__ mord, const float* __restrict__ ssum,
    const float* __restrict__ hmat, float* __restrict__ agg) {
    __shared__ float alpha[H];
    int e = blockIdx.x;
    int s = (e < E) ? src[e] : (e - E);
    int d = (e < E) ? dst[e] : (e - E);
    int t = threadIdx.x;
    if (t < H) {
        float v = asrc[s * H + t] + adst[d * H + t];
        v = v > 0.f ? v : NEG_SLOPE * v;
        float p = __expf(v - ord2flt(mord[d * H + t]));
        alpha[t] = p / (ssum[d * H + t] + 1e-16f);
    }
    __syncthreads();
    float val = hmat[(long)s * (H * D) + t] * alpha[t / D];
    atomicAdd(&agg[(long)d * (H * D) + t], val);
}

// ---------------------------------------------------------------------------
// x = elu(x + bias)  (in place, layer-1 activation)
// ---------------------------------------------------------------------------
__global__ void bias_elu(float* __restrict__ x, const float* __restrict__ bias,
                         int Nn, int C) {
    long i = (long)blockIdx.x * blockDim.x + threadIdx.x;
    if (i >= (long)Nn * C) return;
    float v = x[i] + bias[i % C];
    x[i] = v > 0.f ? v : expm1f(v);
}

// ---------------------------------------------------------------------------
// out = agg + bias (final layer, no activation)
// ---------------------------------------------------------------------------
__global__ void bias_add(const float* __restrict__ x, const float* __restrict__ bias,
                         float* __restrict__ out, int Nn, int C) {
    long i = (long)blockIdx.x * blockDim.x + threadIdx.x;
    if (i >= (long)Nn * C) return;
    out[i] = x[i] + bias[i % C];
}

// ---------------------------------------------------------------------------
extern "C" void kernel_launch(void* const* d_in, const int* in_sizes, int n_in,
                              void* d_out, int out_size, void* d_ws, size_t ws_size,
                              hipStream_t stream) {
    const float* node_emb = (const float*)d_in[0];
    const float* W1       = (const float*)d_in[1];
    const float* att1_s   = (const float*)d_in[2];
    const float* att1_d   = (const float*)d_in[3];
    const float* b1       = (const float*)d_in[4];
    const float* W2       = (const float*)d_in[5];
    const float* att2_s   = (const float*)d_in[6];
    const float* att2_d   = (const float*)d_in[7];
    const float* b2       = (const float*)d_in[8];
    const int*   edge     = (const int*)d_in[9];   // int32 edge_index [2,E]

    const int Nn = in_sizes[0] / EMB;   // 50000
    const int E  = in_sizes[9] / 2;     // 500000
    const int* esrc = edge;
    const int* edst = edge + E;

    // ---- carve workspace (64-float aligned) ----
    float* ws = (float*)d_ws;
    size_t off = 0;
    auto alloc = [&](size_t n) { float* p = ws + off; off += (n + 63) & ~(size_t)63; return p; };
    float*    h1   = alloc((size_t)Nn * HID);
    float*    agg1 = alloc((size_t)Nn * HID);    // becomes x1 after bias+elu
    float*    as1  = alloc((size_t)Nn * HEADS);
    float*    ad1  = alloc((size_t)Nn * HEADS);
    unsigned* m1   = (unsigned*)alloc((size_t)Nn * HEADS);
    float*    s1   = alloc((size_t)Nn * HEADS);
    float*    h2   = alloc((size_t)Nn * REPR);
    float*    agg2 = alloc((size_t)Nn * REPR);
    float*    as2  = alloc((size_t)Nn);
    float*    ad2  = alloc((size_t)Nn);
    unsigned* m2   = (unsigned*)alloc((size_t)Nn);
    float*    s2   = alloc((size_t)Nn);

    const int Etot = E + Nn;            // edges + self-loops

    // ---- init scratch (max/sum/agg buffers) ----
    {
        long total = (long)Nn * HID;
        init_ws<<<dim3((unsigned)((total + 255) / 256)), dim3(256), 0, stream>>>(
            m1, s1, agg1, m2, s2, agg2, Nn);
    }

    // ===================== Layer 1 =====================
    {   // h1 = node_emb @ W1   (50000x128 @ 128x256), 16x64 strip per wave
        int waves = ((Nn + 15) / 16) * (HID / 64);
        gemm_wmma_f32<4, EMB, HID>
            <<<dim3((unsigned)((waves * 32 + 127) / 128)), dim3(128), 0, stream>>>(
            node_emb, W1, h1, Nn);
    }
    att_coef<HEADS, OUT1><<<dim3((Nn * HEADS + 255) / 256), dim3(256), 0, stream>>>(
        h1, att1_s, att1_d, as1, ad1, Nn);
    edge_max<HEADS><<<dim3((Etot * HEADS + 255) / 256), dim3(256), 0, stream>>>(
        esrc, edst, E, Nn, as1, ad1, m1);
    edge_sum<HEADS><<<dim3((Etot * HEADS + 255) / 256), dim3(256), 0, stream>>>(
        esrc, edst, E, Nn, as1, ad1, m1, s1);
    edge_msg<HEADS, OUT1><<<dim3((unsigned)Etot), dim3(HEADS * OUT1), 0, stream>>>(
        esrc, edst, E, Nn, as1, ad1, m1, s1, h1, agg1);
    bias_elu<<<dim3((unsigned)(((long)Nn * HID + 255) / 256)), dim3(256), 0, stream>>>(
        agg1, b1, Nn, HID);

    // ===================== Layer 2 =====================
    {   // h2 = x1 @ W2   (50000x256 @ 256x64), 16x64 strip per wave
        int waves = ((Nn + 15) / 16) * (REPR / 64);
        gemm_wmma_f32<4, HID, REPR>
            <<<dim3((unsigned)((waves * 32 + 127) / 128)), dim3(128), 0, stream>>>(
            agg1, W2, h2, Nn);
    }
    att_coef<1, REPR><<<dim3((Nn + 255) / 256), dim3(256), 0, stream>>>(
        h2, att2_s, att2_d, as2, ad2, Nn);
    edge_max<1><<<dim3((Etot + 255) / 256), dim3(256), 0, stream>>>(
        esrc, edst, E, Nn, as2, ad2, m2);
    edge_sum<1><<<dim3((Etot + 255) / 256), dim3(256), 0, stream>>>(
        esrc, edst, E, Nn, as2, ad2, m2, s2);
    edge_msg<1, REPR><<<dim3((unsigned)Etot), dim3(REPR), 0, stream>>>(
        esrc, edst, E, Nn, as2, ad2, m2, s2, h2, agg2);
    bias_add<<<dim3((unsigned)(((long)Nn * REPR + 255) / 256)), dim3(256), 0, stream>>>(
        agg2, b2, (float*)d_out, Nn, REPR);
}